// ToneStackStage_28810640621781
// MI455X (gfx1250) — compile-verified
//
#include <hip/hip_runtime.h>
#include <hip/hip_bf16.h>

// Problem geometry
#define NB      64
#define LSIG    480000
#define TILE    256               // samples per wave-tile = 16 chunks x 16
#define NTILES  (LSIG / TILE)     // 1875 (exact)
#define NSEC    3
#define CSTRIDE 128               // floats of derived coeffs per (batch,section)
#define FSTRIDE 320               // floats of prebuilt WMMA fragments per (b,sec)

typedef __attribute__((ext_vector_type(2))) float v2f;
typedef __attribute__((ext_vector_type(8))) float v8f;

__device__ __forceinline__ v8f wmma_f32_16x16x4(v2f a, v2f b, v8f c) {
  // 8 args: (neg_a, A, neg_b, B, c_mod, C, reuse_a, reuse_b)
  return __builtin_amdgcn_wmma_f32_16x16x4_f32(false, a, false, b, (short)0, c,
                                               false, false);
}

// ---------------------------------------------------------------------------
// Kernel 0: MLP controller + derived state-space quantities per (batch,section)
//   coef layout (per b,sec; CSTRIDE floats):
//     [16..31] f0[j]  (A^{15-j} B)[0]   [32..47] f1[j]  (chunk in->state)
//     [80..95]  A^16, A^32, A^64, A^128 (row major, scan multipliers)
//     [96..99]  A^256 (row major, cross-tile scan)
//   fr layout (per b,sec; FSTRIDE floats): prebuilt per-lane WMMA A-operands
//     [0..255]   HF: 4 K-fragments x 32 lanes x v2f   (H matrix operand)
//     [256..319] GF: 32 lanes x v2f                   (G16 operand, K-padded)
//   gt: per (b,sec) 256 rows of (C A^n), n = 0..255   (cross-tile fixup)
// ---------------------------------------------------------------------------
__global__ void derive_kernel(const float* __restrict__ knobs,
                              const float* __restrict__ W1,
                              const float* __restrict__ b1v,
                              const float* __restrict__ W2,
                              const float* __restrict__ b2v,
                              float* __restrict__ coef,
                              float* __restrict__ fr,
                              float* __restrict__ gt) {
  int b = threadIdx.x;
  if (b >= NB) return;
  float k0 = knobs[b * 3 + 0], k1 = knobs[b * 3 + 1], k2 = knobs[b * 3 + 2];
  float hmlp[16];
  for (int i = 0; i < 16; ++i)
    hmlp[i] = tanhf(k0 * W1[0 * 16 + i] + k1 * W1[1 * 16 + i] +
                    k2 * W1[2 * 16 + i] + b1v[i]);
  float p[15];
  for (int j = 0; j < 15; ++j) {
    float acc = b2v[j];
    for (int i = 0; i < 16; ++i) acc += hmlp[i] * W2[i * 15 + j];
    p[j] = acc;
  }
  for (int s = 0; s < NSEC; ++s) {
    float c0 = p[s * 5 + 0], c1 = p[s * 5 + 1], c2 = p[s * 5 + 2];
    float a2c = 0.999f * tanhf(p[s * 5 + 4]);
    float a1c = 0.999f * (1.0f + a2c) * tanhf(p[s * 5 + 3]);
    // DF2T state space: s=(s1,s2), A=[[-a1,1],[-a2,0]],
    // B=[b1-a1*b0, b2-a2*b0], y = [1,0]*s_prev + b0*x
    float A00 = -a1c, A01 = 1.0f, A10 = -a2c, A11 = 0.0f;
    float Bv0 = c1 - a1c * c0, Bv1 = c2 - a2c * c0;
    float* cf = coef + (b * NSEC + s) * CSTRIDE;

    float h[16];   // impulse response: h[0]=b0, h[k]=C A^{k-1} B
    {
      float v0 = Bv0, v1 = Bv1;
      h[0] = c0;
      for (int k = 1; k < 16; ++k) {
        h[k] = v0;
        float n0 = A00 * v0 + A01 * v1;
        float n1 = A10 * v0 + A11 * v1;
        v0 = n0; v1 = n1;
      }
    }
    {  // f_j = A^{15-j} B
      float v0 = Bv0, v1 = Bv1;
      cf[16 + 15] = v0; cf[32 + 15] = v1;
      for (int j = 14; j >= 0; --j) {
        float n0 = A00 * v0 + A01 * v1;
        float n1 = A10 * v0 + A11 * v1;
        v0 = n0; v1 = n1;
        cf[16 + j] = v0; cf[32 + j] = v1;
      }
    }
    float g0i[16], g1i[16];  // g_i = C A^i (row vector, C=[1,0])
    {
      float g0 = 1.0f, g1 = 0.0f;
      for (int i = 0; i < 16; ++i) {
        g0i[i] = g0; g1i[i] = g1;
        float n0 = g0 * A00 + g1 * A10;
        float n1 = g0 * A01 + g1 * A11;
        g0 = n0; g1 = n1;
      }
    }
    {  // A^16, A^32, A^64, A^128, A^256 by repeated squaring
      float M00 = A00, M01 = A01, M10 = A10, M11 = A11;
      for (int q = 0; q < 4; ++q) {   // -> A^16
        float n00 = M00 * M00 + M01 * M10, n01 = M00 * M01 + M01 * M11;
        float n10 = M10 * M00 + M11 * M10, n11 = M10 * M01 + M11 * M11;
        M00 = n00; M01 = n01; M10 = n10; M11 = n11;
      }
      for (int q = 0; q < 5; ++q) {   // store A^16,32,64,128,256
        cf[80 + 4 * q + 0] = M00; cf[80 + 4 * q + 1] = M01;
        cf[80 + 4 * q + 2] = M10; cf[80 + 4 * q + 3] = M11;
        float n00 = M00 * M00 + M01 * M10, n01 = M00 * M01 + M01 * M11;
        float n10 = M10 * M00 + M11 * M10, n11 = M10 * M01 + M11 * M11;
        M00 = n00; M01 = n01; M10 = n10; M11 = n11;
      }
    }
    // Prebuilt per-lane WMMA A-operand fragments (branch-free hot path)
    {
      float* fb = fr + (b * NSEC + s) * FSTRIDE;
      for (int lane = 0; lane < 32; ++lane) {
        int m = lane & 15, half = lane >> 4;
        for (int k = 0; k < 4; ++k) {
          int K0 = 4 * k + 2 * half;   // ISA 16x4 f32 A layout K mapping
          fb[(k * 32 + lane) * 2 + 0] = (K0 <= m)     ? h[m - K0]     : 0.0f;
          fb[(k * 32 + lane) * 2 + 1] = (K0 + 1 <= m) ? h[m - K0 - 1] : 0.0f;
        }
        fb[256 + lane * 2 + 0] = half ? 0.0f : g0i[m];  // G16: K=0 col g0
        fb[256 + lane * 2 + 1] = half ? 0.0f : g1i[m];  // K=1 col g1; K=2,3 zero
      }
    }
    {  // Gtile rows: C A^n, n = 0..255
      float g0 = 1.0f, g1 = 0.0f;
      float* G = gt + (b * NSEC + s) * TILE * 2;
      for (int n = 0; n < TILE; ++n) {
        G[n * 2 + 0] = g0; G[n * 2 + 1] = g1;
        float n0 = g0 * A00 + g1 * A10;
        float n1 = g0 * A01 + g1 * A11;
        g0 = n0; g1 = n1;
      }
    }
  }
}

// ---------------------------------------------------------------------------
// Kernel A: one wave per 256-sample tile.
//   Optionally applies the previous section's cross-tile state fixup to its
//   input (fused rank-2 update), then:
//   Y0 = H(16x16) * X(16x16) via 4x v_wmma_f32_16x16x4_f32 (K split),
//   log-depth chunk-state scan, 5th WMMA adds G*S_in fixup,
//   emits tile zero-state out-state to ws.
// ---------------------------------------------------------------------------
__global__ void __launch_bounds__(256) tile_kernel(const float* in, float* out,
                                                   const float* __restrict__ coef,
                                                   const float* __restrict__ fr,
                                                   const float* __restrict__ gt,
                                                   const float* __restrict__ sg,
                                                   float* __restrict__ ts,
                                                   int sec, int prevsec) {
  __shared__ float lds[8 * TILE];
  const int lane = threadIdx.x & 31;
  const int wave = threadIdx.x >> 5;
  const int gw = blockIdx.x * 8 + wave;       // exactly NB*NTILES waves
  const int b = gw / NTILES;
  const int t = gw - b * NTILES;
  const int base = b * LSIG + t * TILE;
  float* xs = lds + wave * TILE;

  // load tile, coalesced 128-bit
  float4 u0 = *(const float4*)(in + base + lane * 8);
  float4 u1 = *(const float4*)(in + base + lane * 8 + 4);

  // fused fixup from previous section: x[n] += (C A^n) . sigma_tile
  if (prevsec >= 0) {
    float2 sv = ((const float2*)sg)[gw];       // uniform per wave
    const float4* G4 = (const float4*)(gt + (b * NSEC + prevsec) * TILE * 2);
    float4 ga = G4[lane * 4 + 0];              // (g0,g1) pairs, n = lane*8+..
    float4 gb = G4[lane * 4 + 1];
    float4 gc = G4[lane * 4 + 2];
    float4 gd = G4[lane * 4 + 3];
    u0.x += ga.x * sv.x + ga.y * sv.y;
    u0.y += ga.z * sv.x + ga.w * sv.y;
    u0.z += gb.x * sv.x + gb.y * sv.y;
    u0.w += gb.z * sv.x + gb.w * sv.y;
    u1.x += gc.x * sv.x + gc.y * sv.y;
    u1.y += gc.z * sv.x + gc.w * sv.y;
    u1.z += gd.x * sv.x + gd.y * sv.y;
    u1.w += gd.z * sv.x + gd.w * sv.y;
  }
  *(float4*)(xs + lane * 8)     = u0;
  *(float4*)(xs + lane * 8 + 4) = u1;
  __syncthreads();

  const float* cf = coef + (b * NSEC + sec) * CSTRIDE;
  const v2f* hf = (const v2f*)(fr + (b * NSEC + sec) * FSTRIDE);
  const int m = lane & 15;     // A: row M / B: col N (chunk index)
  const int half = lane >> 4;  // lane-group -> K pair selection

  // Load this lane's column of X once: X[j][n] = xs[n*16 + j], n = m
  float4 xc[4];
#pragma unroll
  for (int q = 0; q < 4; ++q)
    xc[q] = ((const float4*)(xs + m * 16))[q];   // 4x ds_load_b128

  // Y0 = H * X  (4 chained WMMAs over the K dimension)
  v8f acc = {0.f, 0.f, 0.f, 0.f, 0.f, 0.f, 0.f, 0.f};
#pragma unroll
  for (int k = 0; k < 4; ++k) {
    v2f a = hf[k * 32 + lane];                       // prebuilt H fragment
    v2f bb;
    bb.x = half ? xc[k].z : xc[k].x;                 // X[K0][n]
    bb.y = half ? xc[k].w : xc[k].y;                 // X[K0+1][n]
    acc = wmma_f32_16x16x4(a, bb, acc);
  }

  // fX_c = F * X_c  for chunk c = m (register-only; lanes 16..31 duplicate)
  float fx0 = 0.f, fx1 = 0.f;
#pragma unroll
  for (int q = 0; q < 4; ++q) {
    fx0 += cf[16 + 4 * q + 0] * xc[q].x + cf[16 + 4 * q + 1] * xc[q].y +
           cf[16 + 4 * q + 2] * xc[q].z + cf[16 + 4 * q + 3] * xc[q].w;
    fx1 += cf[32 + 4 * q + 0] * xc[q].x + cf[32 + 4 * q + 1] * xc[q].y +
           cf[32 + 4 * q + 2] * xc[q].z + cf[32 + 4 * q + 3] * xc[q].w;
  }

  // Log-depth chunk-state scan: chunk maps v -> A^16 v + fX_c have a uniform
  // linear part, so Hillis-Steele only propagates the 2-vector parts with
  // multipliers A^16, A^32, A^64, A^128 (4 steps instead of 16).
  float p0 = fx0, p1 = fx1;            // inclusive prefix (window length 1)
#pragma unroll
  for (int s = 0; s < 4; ++s) {
    float M00 = cf[80 + 4 * s + 0], M01 = cf[80 + 4 * s + 1];
    float M10 = cf[80 + 4 * s + 2], M11 = cf[80 + 4 * s + 3];
    int len = 1 << s;
    float q0 = __shfl_up(p0, (unsigned)len, 32);
    float q1 = __shfl_up(p1, (unsigned)len, 32);
    if (m >= len) {
      p0 = M00 * q0 + M01 * q1 + p0;
      p1 = M10 * q0 + M11 * q1 + p1;
    }
  }
  // exclusive prefix = incoming state s_in for this lane's chunk
  float e0 = __shfl_up(p0, 1u, 32);
  float e1 = __shfl_up(p1, 1u, 32);
  float s0c = (m >= 1) ? e0 : 0.f;
  float s1c = (m >= 1) ? e1 : 0.f;

  // Y += G16(16x2, K-padded) * S_in(2x16, K-padded)  -- 5th WMMA
  {
    v2f a = hf[128 + lane];          // prebuilt G fragment (zeros for K=2,3)
    v2f bb;
    bb.x = half ? 0.f : s0c;         // S[0][n]
    bb.y = half ? 0.f : s1c;         // S[1][n]
    acc = wmma_f32_16x16x4(a, bb, acc);
  }

  if (lane == 15) {  // inclusive prefix at chunk 15 = tile zero-state out-state
    ts[(b * NTILES + t) * 2 + 0] = p0;
    ts[(b * NTILES + t) * 2 + 1] = p1;
  }

  // stage D back through LDS (two b128 stores), then coalesced global b128
  __syncthreads();
  {
    float4 d0, d1;                       // sample index = n*16 + M, M contig.
    d0.x = acc[0]; d0.y = acc[1]; d0.z = acc[2]; d0.w = acc[3];
    d1.x = acc[4]; d1.y = acc[5]; d1.z = acc[6]; d1.w = acc[7];
    *(float4*)(xs + m * 16 + 8 * half)     = d0;
    *(float4*)(xs + m * 16 + 8 * half + 4) = d1;
  }
  __syncthreads();
  *(float4*)(out + base + lane * 8)     = *(const float4*)(xs + lane * 8);
  *(float4*)(out + base + lane * 8 + 4) = *(const float4*)(xs + lane * 8 + 4);
}

// ---------------------------------------------------------------------------
// Kernel B: per batch row, serial scan of tile states: sigma_{t+1} = A^256 sigma_t + z_t
// ---------------------------------------------------------------------------
__global__ void scan_kernel(const float* __restrict__ coef,
                            const float* __restrict__ ts,
                            float* __restrict__ sg, int sec) {
  int b = threadIdx.x;
  if (b >= NB) return;
  const float* cf = coef + (b * NSEC + sec) * CSTRIDE;
  float A00 = cf[96], A01 = cf[97], A10 = cf[98], A11 = cf[99];  // A^256
  float s0 = 0.f, s1 = 0.f;
#pragma unroll 5
  for (int t = 0; t < NTILES; ++t) {
    int idx = (b * NTILES + t) * 2;
    sg[idx + 0] = s0; sg[idx + 1] = s1;   // incoming state for tile t
    float z0 = ts[idx + 0], z1 = ts[idx + 1];
    float n0 = A00 * s0 + A01 * s1 + z0;
    float n1 = A10 * s0 + A11 * s1 + z1;
    s0 = n0; s1 = n1;
  }
}

// ---------------------------------------------------------------------------
// Kernel C: y[n] += (C A^n) . sigma_tile  (standalone fixup, last section only)
// ---------------------------------------------------------------------------
__global__ void __launch_bounds__(256) fixup_kernel(float* out,
                                                    const float* __restrict__ gt,
                                                    const float* __restrict__ sg,
                                                    int sec) {
  int tile = blockIdx.x;        // = b*NTILES + t
  int b = tile / NTILES;
  int t = tile - b * NTILES;
  int n = threadIdx.x;
  float s0 = sg[tile * 2 + 0], s1 = sg[tile * 2 + 1];
  const float2* G = (const float2*)(gt + (b * NSEC + sec) * TILE * 2);
  float2 g = G[n];
  int idx = b * LSIG + t * TILE + n;
  out[idx] += g.x * s0 + g.y * s1;
}

// ---------------------------------------------------------------------------
extern "C" void kernel_launch(void* const* d_in, const int* in_sizes, int n_in,
                              void* d_out, int out_size, void* d_ws, size_t ws_size,
                              hipStream_t stream) {
  (void)in_sizes; (void)n_in; (void)out_size; (void)ws_size;
  const float* x     = (const float*)d_in[0];
  const float* knobs = (const float*)d_in[1];
  const float* W1    = (const float*)d_in[2];
  const float* b1v   = (const float*)d_in[3];
  const float* W2    = (const float*)d_in[4];
  const float* b2v   = (const float*)d_in[5];
  float* out = (float*)d_out;

  // workspace carve-up (~2.7 MB total)
  float* COEF = (float*)d_ws;                       // 64*3*128
  float* FR   = COEF + NB * NSEC * CSTRIDE;         // 64*3*320
  float* GT   = FR   + NB * NSEC * FSTRIDE;         // 64*3*512
  float* TS   = GT   + NB * NSEC * TILE * 2;        // 64*1875*2
  float* SG   = TS   + NB * NTILES * 2;             // 64*1875*2

  derive_kernel<<<1, 64, 0, stream>>>(knobs, W1, b1v, W2, b2v, COEF, FR, GT);

  const int tileBlocks = (NB * NTILES) / 8;  // 8 waves/block, exact
  for (int sec = 0; sec < NSEC; ++sec) {
    const float* src = (sec == 0) ? x : out;
    // sections 1,2 fuse the previous section's cross-tile fixup into the load
    tile_kernel<<<tileBlocks, 256, 0, stream>>>(src, out, COEF, FR, GT, SG, TS,
                                                sec, sec - 1);
    scan_kernel<<<1, 64, 0, stream>>>(COEF, TS, SG, sec);
  }
  // last section still needs its standalone state fixup
  fixup_kernel<<<NB * NTILES, 256, 0, stream>>>(out, GT, SG, NSEC - 1);
}